// ModelStep_57715770524051
// MI455X (gfx1250) — compile-verified
//
#include <hip/hip_runtime.h>
#include <math.h>

// ---------------------------------------------------------------------------
// MI455X (gfx1250) batched DOPRI5 (jax.experimental.ode.odeint semantics)
// for the 11-species RKIP mass-action network.
//
// Compute-bound fp32 VALU workload: state + all 7 RK stages live in VGPRs,
// dy/dt = S@v hand-unrolled (S is +/-1 sparse, 23 adds). Input staged per-wave
// into LDS by the CDNA5 Tensor Data Mover (tensor_load_to_lds + tensorcnt
// wait); lane reads are dword-stride-11 => bank-conflict free (gcd(11,64)=1).
// WMMA intentionally unused: the 11x11 +/-1-sparse matvec is 2-3x faster as
// unrolled VALU adds (32 traj/wave) than a padded 16x16x4 f32 WMMA scheme
// (16 traj/wave + cross-half lane marshaling per eval).
// ---------------------------------------------------------------------------

#define NSP 11
#define THREADS 256
#define WSZ 32
#define PER_WAVE_F (WSZ * NSP)      // 352 floats = 1408 bytes per wave
#define TILE_F (THREADS * NSP)      // 2816 floats per block

#define RTOLC 1e-6f
#define ATOLC 1e-6f
#define MXSTEP 1000

typedef __attribute__((ext_vector_type(4))) unsigned int v4u;
typedef __attribute__((ext_vector_type(8))) int v8i;
typedef __attribute__((ext_vector_type(4))) int v4i;

// ---- jax DOPRI5 tableau (computed in double, stored fp32) ------------------
__device__ __constant__ static const float kB21 = (float)(1.0/5.0);
__device__ __constant__ static const float kB31 = (float)(3.0/40.0),  kB32 = (float)(9.0/40.0);
__device__ __constant__ static const float kB41 = (float)(44.0/45.0), kB42 = (float)(-56.0/15.0), kB43 = (float)(32.0/9.0);
__device__ __constant__ static const float kB51 = (float)(19372.0/6561.0), kB52 = (float)(-25360.0/2187.0),
                                           kB53 = (float)(64448.0/6561.0), kB54 = (float)(-212.0/729.0);
__device__ __constant__ static const float kB61 = (float)(9017.0/3168.0),  kB62 = (float)(-355.0/33.0),
                                           kB63 = (float)(46732.0/5247.0), kB64 = (float)(49.0/176.0),
                                           kB65 = (float)(-5103.0/18656.0);
// c_sol (row 7 of beta == c_sol, FSAL)
__device__ __constant__ static const float kC1 = (float)(35.0/384.0), kC3 = (float)(500.0/1113.0),
                                           kC4 = (float)(125.0/192.0), kC5 = (float)(-2187.0/6784.0),
                                           kC6 = (float)(11.0/84.0);
// c_error (jax values)
__device__ __constant__ static const float kE1 = (float)(35.0/384.0 - 1951.0/21600.0);
__device__ __constant__ static const float kE3 = (float)(500.0/1113.0 - 22642.0/50085.0);
__device__ __constant__ static const float kE4 = (float)(125.0/192.0 - 451.0/720.0);
__device__ __constant__ static const float kE5 = (float)(-2187.0/6784.0 + 12231.0/42400.0);
__device__ __constant__ static const float kE6 = (float)(11.0/84.0 - 649.0/6300.0);
__device__ __constant__ static const float kE7 = (float)(-1.0/60.0);
// dps_c_mid (for interp_fit_dopri)
__device__ __constant__ static const float kM1 = (float)(6025192743.0/30085553152.0/2.0);
__device__ __constant__ static const float kM3 = (float)(51252292925.0/65400821598.0/2.0);
__device__ __constant__ static const float kM4 = (float)(-2691868925.0/45128329728.0/2.0);
__device__ __constant__ static const float kM5 = (float)(187940372067.0/1594534317056.0/2.0);
__device__ __constant__ static const float kM6 = (float)(-1776094331.0/19743644256.0/2.0);
__device__ __constant__ static const float kM7 = (float)(11237099.0/235043384.0/2.0);

// ---- dy/dt: v = rates(y), dy = S @ v  (S hand-unrolled, +/-1 sparse) -------
__device__ __forceinline__ void deriv(const float* __restrict__ K,
                                      const float* __restrict__ y,
                                      float* __restrict__ dy) {
  const float v0  = K[0]  * y[0] * y[1];
  const float v1  = K[1]  * y[2];
  const float v2  = K[2]  * y[2] * y[3];
  const float v3  = K[3]  * y[4];
  const float v4  = K[4]  * y[4];
  const float v5  = K[5]  * y[5] * y[7];
  const float v6  = K[6]  * y[8];
  const float v7  = K[7]  * y[8];
  const float v8  = K[8]  * y[6] * y[9];
  const float v9  = K[9]  * y[10];
  const float v10 = K[10] * y[10];
  dy[0]  = v1 + v4  - v0;
  dy[1]  = v1 + v10 - v0;
  dy[2]  = (v0 - v1) + (v3 - v2);
  dy[3]  = v3 + v7  - v2;
  dy[4]  = v2 - v3 - v4;
  dy[5]  = v4 + v6 - v5;
  dy[6]  = v4 + v9 - v8;
  dy[7]  = v6 + v7 - v5;
  dy[8]  = v5 - v6 - v7;
  dy[9]  = v9 + v10 - v8;
  dy[10] = v8 - v9 - v10;
}

__global__ __launch_bounds__(THREADS) void rkip_dopri5_kernel(
    const float* __restrict__ y_in,
    const float* __restrict__ c_in,
    const float* __restrict__ t_in,
    const float* __restrict__ dT_in,
    float* __restrict__ out,
    int batch) {
  __shared__ float tile[TILE_F];
  const int tid = (int)threadIdx.x;
  const int wid = tid >> 5;
  const size_t blk0 = (size_t)blockIdx.x * THREADS;
  const size_t gidx = blk0 + (size_t)tid;
  const bool fullblock = (blk0 + THREADS) <= (size_t)batch;

#if __has_builtin(__builtin_amdgcn_tensor_load_to_lds) && __has_builtin(__builtin_amdgcn_s_wait_tensorcnt)
#define USE_TDM 1
#else
#define USE_TDM 0
#endif

  // ---- issue per-wave TDM DMA of this wave's 32 input rows (no wait yet) ---
  if (fullblock) {
#if USE_TDM
    const unsigned swid = (unsigned)__builtin_amdgcn_readfirstlane(wid);
    const unsigned lds_base =
        (unsigned)(size_t)(&tile[0]) + swid * (unsigned)(PER_WAVE_F * 4);
    const unsigned long long ga =
        (unsigned long long)(size_t)y_in +
        ((unsigned long long)blk0 + (unsigned long long)swid * WSZ) * (NSP * 4ull);
    v4u g0; v8i g1; v4i gz4; v8i gz8;
    // D# group0: count=1 | lds_addr | global_addr[56:0] | type=2
    g0[0] = 1u;
    g0[1] = lds_base;
    g0[2] = (unsigned)(ga & 0xFFFFFFFFull);
    g0[3] = (unsigned)((ga >> 32) & 0x01FFFFFFull) | (2u << 30);
    // D# group1: data_size=4B; tensor_dim0=352, tensor_dim1=1;
    //            tile_dim0=352, tile_dim1=1; dim0_stride=352
    g1[0] = (int)(2u << 16);                      // data_size=2 (4 bytes)
    g1[1] = (int)((unsigned)PER_WAVE_F << 16);    // tensor_dim0[15:0]
    g1[2] = (int)(1u << 16);                      // tensor_dim0 hi=0 | tensor_dim1=1
    g1[3] = (int)((unsigned)PER_WAVE_F << 16);    // tensor_dim1 hi=0 | tile_dim0
    g1[4] = 1;                                    // tile_dim1=1 | tile_dim2=0
    g1[5] = PER_WAVE_F;                           // tensor_dim0_stride lo32
    g1[6] = 0;
    g1[7] = 0;
    gz4[0] = 0; gz4[1] = 0; gz4[2] = 0; gz4[3] = 0;  // groups 2/3 unused (<=2D)
    gz8[0] = 0; gz8[1] = 0; gz8[2] = 0; gz8[3] = 0;
    gz8[4] = 0; gz8[5] = 0; gz8[6] = 0; gz8[7] = 0;
    __builtin_amdgcn_tensor_load_to_lds(g0, g1, gz4, gz4, gz8, 0);
#else
    {
      const int lane = tid & (WSZ - 1);
      const size_t wbase = blk0 * NSP + (size_t)wid * PER_WAVE_F;
      #pragma unroll
      for (int i = 0; i < NSP; ++i)
        tile[wid * PER_WAVE_F + lane + i * WSZ] = y_in[wbase + lane + i * WSZ];
      __syncthreads();
    }
#endif
  }

  if (gidx < (size_t)batch) {
    // ---- uniform constants load overlaps the TDM DMA -----------------------
    float K[NSP];
    const float vol = c_in[NSP];
    #pragma unroll
    for (int i = 0; i < NSP; ++i) K[i] = c_in[i] * vol;
    const float t0 = t_in[0];
    const float dT = dT_in[0];
    const float target = t0 + dT;

    float y[NSP];
    if (fullblock) {
#if USE_TDM
      __builtin_amdgcn_s_wait_tensorcnt(0);
      __asm__ __volatile__("" ::: "memory");
#endif
      // stride-11 dword LDS reads: conflict-free (gcd(11,64)==1)
      #pragma unroll
      for (int s = 0; s < NSP; ++s) y[s] = tile[tid * NSP + s];
    } else {
      #pragma unroll
      for (int s = 0; s < NSP; ++s) y[s] = y_in[gidx * NSP + s];
    }

    float f[NSP];
    deriv(K, y, f);

    // ---- initial step size (Hairer II.4, order=4 like jax) -----------------
    float dt;
    {
      float scale[NSP], d0s = 0.f, d1s = 0.f;
      #pragma unroll
      for (int s = 0; s < NSP; ++s) {
        scale[s] = ATOLC + fabsf(y[s]) * RTOLC;
        const float inv = __builtin_amdgcn_rcpf(scale[s]);
        const float a = y[s] * inv; d0s += a * a;
        const float b = f[s] * inv; d1s += b * b;
      }
      const float d0 = sqrtf(d0s), d1 = sqrtf(d1s);
      const float h0 = (d0 < 1e-5f || d1 < 1e-5f) ? 1e-6f : 0.01f * d0 / d1;
      float yt[NSP], ft[NSP];
      #pragma unroll
      for (int s = 0; s < NSP; ++s) yt[s] = y[s] + h0 * f[s];
      deriv(K, yt, ft);
      float d2s = 0.f;
      #pragma unroll
      for (int s = 0; s < NSP; ++s) {
        const float a = (ft[s] - f[s]) * __builtin_amdgcn_rcpf(scale[s]);
        d2s += a * a;
      }
      const float d2 = sqrtf(d2s) / h0;
      const float h1 = (d1 <= 1e-15f && d2 <= 1e-15f)
                           ? fmaxf(1e-6f, h0 * 1e-3f)
                           : powf(0.01f / fmaxf(d1, d2), 0.2f);
      dt = fminf(100.f * h0, h1);
    }

    // ---- adaptive DOPRI5 loop (jax while_loop semantics) -------------------
    float t = t0, last_t = t0, hs = 0.f;
    float yp[NSP], ym[NSP], f0s[NSP];   // accepted-step interp state
    #pragma unroll
    for (int s = 0; s < NSP; ++s) { yp[s] = y[s]; ym[s] = y[s]; f0s[s] = f[s]; }

    int steps = 0;
    while (t < target && steps < MXSTEP && dt > 0.f) {
      ++steps;
      float yt[NSP], k2[NSP], k3[NSP], k4[NSP], k5[NSP], k6[NSP], k7[NSP], y1[NSP];
      #pragma unroll
      for (int s = 0; s < NSP; ++s) yt[s] = y[s] + dt * (kB21 * f[s]);
      deriv(K, yt, k2);
      #pragma unroll
      for (int s = 0; s < NSP; ++s) yt[s] = y[s] + dt * (kB31 * f[s] + kB32 * k2[s]);
      deriv(K, yt, k3);
      #pragma unroll
      for (int s = 0; s < NSP; ++s) yt[s] = y[s] + dt * (kB41 * f[s] + kB42 * k2[s] + kB43 * k3[s]);
      deriv(K, yt, k4);
      #pragma unroll
      for (int s = 0; s < NSP; ++s)
        yt[s] = y[s] + dt * (kB51 * f[s] + kB52 * k2[s] + kB53 * k3[s] + kB54 * k4[s]);
      deriv(K, yt, k5);
      #pragma unroll
      for (int s = 0; s < NSP; ++s)
        yt[s] = y[s] + dt * (kB61 * f[s] + kB62 * k2[s] + kB63 * k3[s] + kB64 * k4[s] + kB65 * k5[s]);
      deriv(K, yt, k6);
      #pragma unroll
      for (int s = 0; s < NSP; ++s)   // stage-7 input == y1 (FSAL)
        y1[s] = y[s] + dt * (kC1 * f[s] + kC3 * k3[s] + kC4 * k4[s] + kC5 * k5[s] + kC6 * k6[s]);
      deriv(K, y1, k7);

      // mean_error_ratio^2 = mean((err/tol)^2); accept iff <= 1
      float errsum = 0.f;
      #pragma unroll
      for (int s = 0; s < NSP; ++s) {
        const float e = dt * (kE1 * f[s] + kE3 * k3[s] + kE4 * k4[s] +
                              kE5 * k5[s] + kE6 * k6[s] + kE7 * k7[s]);
        const float tol = ATOLC + RTOLC * fmaxf(fabsf(y[s]), fabsf(y1[s]));
        const float r = e * __builtin_amdgcn_rcpf(tol);
        errsum += r * r;
      }
      const float r2m = errsum * (1.0f / (float)NSP);  // = ratio^2

      // optimal_step_size: safety=0.9, ifactor=10, dfactor=0.2 (->1 if accept)
      // ratio^(-1/5) == (r2m)^(-1/10)
      float ndt;
      if (r2m == 0.f) {
        ndt = dt * 10.f;
      } else {
        const float dfac = (r2m < 1.f) ? 1.f : 0.2f;
        const float fac = fminf(10.f, fmaxf(0.9f * exp2f(-0.1f * log2f(r2m)), dfac));
        ndt = dt * fac;
      }
      if (r2m <= 1.f) {   // accept: save interp state, advance
        #pragma unroll
        for (int s = 0; s < NSP; ++s) {
          const float ymid = y[s] + dt * (kM1 * f[s] + kM3 * k3[s] + kM4 * k4[s] +
                                          kM5 * k5[s] + kM6 * k6[s] + kM7 * k7[s]);
          yp[s] = y[s]; f0s[s] = f[s]; ym[s] = ymid;
          y[s] = y1[s]; f[s] = k7[s];
        }
        hs = dt; last_t = t; t = t + dt;
      }
      dt = fmaxf(ndt, 0.f);
    }

    // ---- interp_fit_dopri quartic, evaluated at target ---------------------
    float res[NSP];
    if (t > last_t) {
      const float sf = (target - last_t) / (t - last_t);
      #pragma unroll
      for (int s = 0; s < NSP; ++s) {
        const float hf0 = hs * f0s[s];
        const float hf1 = hs * f[s];
        const float A = -2.f * hf0 + 2.f * hf1 - 8.f * yp[s] - 8.f * y[s] + 16.f * ym[s];
        const float B =  5.f * hf0 - 3.f * hf1 + 18.f * yp[s] + 14.f * y[s] - 32.f * ym[s];
        const float C = -4.f * hf0 +       hf1 - 11.f * yp[s] -  5.f * y[s] + 16.f * ym[s];
        res[s] = (((A * sf + B) * sf + C) * sf + hf0) * sf + yp[s];
      }
    } else {
      #pragma unroll
      for (int s = 0; s < NSP; ++s) res[s] = y[s];
    }

    float* __restrict__ dst = out + gidx * (size_t)NSP;
    #pragma unroll
    for (int s = 0; s < NSP; ++s) dst[s] = res[s];
  }

  // ---- tail outputs: c echo (12) + t_new (1) -------------------------------
  if (blockIdx.x == 0 && tid < 13) {
    float* tail = out + (size_t)batch * NSP;
    tail[tid] = (tid < 12) ? c_in[tid] : (t_in[0] + dT_in[0]);
  }
}

extern "C" void kernel_launch(void* const* d_in, const int* in_sizes, int n_in,
                              void* d_out, int out_size, void* d_ws, size_t ws_size,
                              hipStream_t stream) {
  (void)n_in; (void)out_size; (void)d_ws; (void)ws_size;
  const float* y  = (const float*)d_in[0];
  /* d_in[1] is w: zero-length, identity passthrough */
  const float* c  = (const float*)d_in[2];
  const float* t  = (const float*)d_in[3];
  const float* dT = (const float*)d_in[4];
  const int batch = in_sizes[0] / NSP;
  float* out = (float*)d_out;
  const int blocks = (batch + THREADS - 1) / THREADS;
  rkip_dopri5_kernel<<<blocks, THREADS, 0, stream>>>(y, c, t, dT, out, batch);
}